// ToRGB_79267916415077
// MI455X (gfx1250) — compile-verified
//
#include <hip/hip_runtime.h>

// ---- problem constants (from the reference) ----
#define IN_CH   128
#define W_DIM   512
#define OUT_CH  3
#define BATCH   8
#define HW      (512 * 512)
#define STYLE_WSCALE 0.04419417382415922f   // 1/sqrt(512)
#define CONV_WSCALE  0.08838834764831845f   // 1/sqrt(128)
#define NEG_SLOPE    0.01f

typedef __attribute__((ext_vector_type(2))) float v2f;
typedef __attribute__((ext_vector_type(4))) float v4f;
typedef __attribute__((ext_vector_type(8))) float v8f;

__device__ __forceinline__ float leaky(float v) {
    return v > 0.0f ? v : NEG_SLOPE * v;
}

// --------------------------------------------------------------------------
// Stage 1: style projection via V_WMMA_F32_16X16X4_F32.
//   s[b][c] = LeakyReLU( sum_k w[b][k] * style_W[c][k] * STYLE_WSCALE + style_b[c] )
// One workgroup, 8 waves. Wave `wv` owns channel tile [16*wv, 16*wv+16).
// A (16x4, MxK): lanes 0-15 hold M=lane (K=k0,k0+1 in v0,v1); lanes 16-31 hold
// K=k0+2,k0+3.  Rows M=8..15 are duplicates of 0..7 (only D rows 0..7 are kept,
// and those live in lanes 0-15 of the 8 accumulator VGPRs).
// B (4x16, KxN): mirrored layout, N=lane&15, same K sub-offset per half-wave.
// --------------------------------------------------------------------------
__launch_bounds__(256)
__global__ void style_wmma_kernel(const float* __restrict__ w,
                                  const float* __restrict__ style_W,
                                  const float* __restrict__ style_b,
                                  float* __restrict__ s_out) {
    const int lane     = threadIdx.x & 31;
    const int wave     = threadIdx.x >> 5;        // 0..7 -> channel tile
    const int chanBase = wave * 16;
    const int m        = lane & 7;                // batch row (8..15 duplicated)
    const int n        = lane & 15;               // channel column in tile
    const int koff     = (lane >> 4) * 2;         // per-half-wave K sub-offset

    const float* __restrict__ arow = w + (size_t)m * W_DIM;
    const float* __restrict__ brow = style_W + (size_t)(chanBase + n) * W_DIM;

    v8f c = {0.f, 0.f, 0.f, 0.f, 0.f, 0.f, 0.f, 0.f};
#pragma unroll 4
    for (int k0 = 0; k0 < W_DIM; k0 += 4) {
        v2f a, b;
        a.x = arow[k0 + koff];
        a.y = arow[k0 + koff + 1];
        b.x = brow[k0 + koff];
        b.y = brow[k0 + koff + 1];
        c = __builtin_amdgcn_wmma_f32_16x16x4_f32(
                /*neg_a=*/false, a, /*neg_b=*/false, b,
                /*c_mod=*/(short)0, c, /*reuse_a=*/false, /*reuse_b=*/false);
    }

    // D layout: VGPR j, lanes 0-15 -> (batch=j, chan=chanBase+lane). Lanes 16-31
    // hold batch 8..15 (garbage duplicates) -> discarded.
    if (lane < 16) {
        const int   chan = chanBase + n;
        const float bb   = style_b[chan];
#pragma unroll
        for (int j = 0; j < 8; ++j) {
            float v = c[j] * STYLE_WSCALE + bb;
            s_out[(size_t)j * IN_CH + chan] = leaky(v);
        }
    }
}

// --------------------------------------------------------------------------
// Stage 2: streaming modulated 1x1 conv (memory bound: ~1.07 GB read @ 23.3 TB/s
// => ~47 us floor; 1.6 GFLOP of math is free).  4 pixels/thread via b128
// non-temporal loads, per-batch effective weights staged once in LDS.
// --------------------------------------------------------------------------
__launch_bounds__(256)
__global__ void torgb_stream_kernel(const float* __restrict__ x,
                                    const float* __restrict__ conv_W,
                                    const float* __restrict__ bias,
                                    const float* __restrict__ s,
                                    float* __restrict__ out) {
    __shared__ v4f weff[IN_CH];        // (w0,w1,w2,pad) per channel -> ds_load_b128

    const int b   = blockIdx.y;
    const int tid = threadIdx.x;

    if (tid < IN_CH) {
        const float sv = s[(size_t)b * IN_CH + tid] * CONV_WSCALE;
        v4f wv;
        wv.x = sv * conv_W[0 * IN_CH + tid];
        wv.y = sv * conv_W[1 * IN_CH + tid];
        wv.z = sv * conv_W[2 * IN_CH + tid];
        wv.w = 0.0f;
        weff[tid] = wv;
    }
    __syncthreads();

    const size_t pix0 = ((size_t)blockIdx.x * blockDim.x + tid) * 4;
    const float* __restrict__ xb = x + (size_t)b * IN_CH * HW + pix0;

    v4f a0 = {0.f, 0.f, 0.f, 0.f};
    v4f a1 = a0, a2 = a0;

#pragma unroll 4
    for (int ci = 0; ci < IN_CH; ++ci) {
        const v4f xv = __builtin_nontemporal_load(
                           (const v4f*)(xb + (size_t)ci * HW));
        const v4f wv = weff[ci];
        a0 += xv * wv.x;
        a1 += xv * wv.y;
        a2 += xv * wv.z;
    }

    const float b0 = bias[0], b1 = bias[1], b2 = bias[2];
    v4f r0, r1, r2;
    r0.x = leaky(a0.x + b0); r0.y = leaky(a0.y + b0);
    r0.z = leaky(a0.z + b0); r0.w = leaky(a0.w + b0);
    r1.x = leaky(a1.x + b1); r1.y = leaky(a1.y + b1);
    r1.z = leaky(a1.z + b1); r1.w = leaky(a1.w + b1);
    r2.x = leaky(a2.x + b2); r2.y = leaky(a2.y + b2);
    r2.z = leaky(a2.z + b2); r2.w = leaky(a2.w + b2);

    float* __restrict__ ob = out + (size_t)b * OUT_CH * HW + pix0;
    __builtin_nontemporal_store(r0, (v4f*)(ob + 0 * (size_t)HW));
    __builtin_nontemporal_store(r1, (v4f*)(ob + 1 * (size_t)HW));
    __builtin_nontemporal_store(r2, (v4f*)(ob + 2 * (size_t)HW));
}

extern "C" void kernel_launch(void* const* d_in, const int* in_sizes, int n_in,
                              void* d_out, int out_size, void* d_ws, size_t ws_size,
                              hipStream_t stream) {
    (void)in_sizes; (void)n_in; (void)out_size; (void)ws_size;
    const float* x       = (const float*)d_in[0];   // [8,128,512,512]
    const float* w       = (const float*)d_in[1];   // [8,512]
    const float* style_W = (const float*)d_in[2];   // [128,512]
    const float* style_b = (const float*)d_in[3];   // [128]
    const float* conv_W  = (const float*)d_in[4];   // [3,128]
    const float* bias    = (const float*)d_in[5];   // [3]
    float*       out     = (float*)d_out;           // [8,3,512,512]
    float*       s_ws    = (float*)d_ws;            // [8,128] scratch

    // Stage 1: 8 waves, one 16x16 WMMA tile each (128 chained wmma per wave).
    style_wmma_kernel<<<dim3(1), dim3(256), 0, stream>>>(w, style_W, style_b, s_ws);

    // Stage 2: 256 thr * 4 px = 1024 px/block; HW/1024 = 256 blocks per batch.
    dim3 grid(HW / (256 * 4), BATCH);
    torgb_stream_kernel<<<grid, dim3(256), 0, stream>>>(x, conv_W, bias, s_ws, out);
}